// ModulatedConv2D_7713761263747
// MI455X (gfx1250) — compile-verified
//
#include <hip/hip_runtime.h>

// MI455X / gfx1250 modulated conv2d (StyleGAN2):
//   out = sigma[b,o] * conv3x3( x * style[b,i], GAIN * weight )
// Implicit GEMM on V_WMMA_F32_16X16X4_F32 (native f32 matrix pipe, wave32).
// Operand tiles are streamed into LDS by the Tensor Data Mover (TDM,
// tensor_load_to_lds) with hardware LDS padding producing bank-conflict-free
// fragment layouts. Hot loop = ds_load fragments + WMMA only.

typedef float v2f __attribute__((ext_vector_type(2)));
typedef float v8f __attribute__((ext_vector_type(8)));
typedef unsigned int u32x4 __attribute__((ext_vector_type(4)));
typedef int i32x4 __attribute__((ext_vector_type(4)));
typedef int i32x8 __attribute__((ext_vector_type(8)));

static constexpr int   NB    = 16;
static constexpr int   IC    = 512;
static constexpr int   OC    = 512;
static constexpr int   HH    = 32;
static constexpr int   WW    = 32;
static constexpr int   HW    = HH * WW;       // 1024
static constexpr int   KTAPS = 9;
static constexpr int   KTOT  = IC * KTAPS;    // 4608, k = tap*512 + i
static constexpr int   PH    = 34;
static constexpr int   PW    = 34;
static constexpr int   PHW   = PH * PW;       // 1156
static constexpr float GAIN  = 0.014731391274719739f; // 1/sqrt(512*9)
static constexpr float EPS   = 1e-8f;

// workspace layout (floats): xs | wtr | wsq | sigma   (~48.4 MB)
static constexpr size_t XS_N  = (size_t)NB * IC * PHW;
static constexpr size_t WT_N  = (size_t)OC * KTOT;
static constexpr size_t WSQ_N = (size_t)OC * IC;

// ------------------------------------------------------------- prep: weight
__global__ __launch_bounds__(256) void prep_w_kernel(const float* __restrict__ w,
                                                     float* __restrict__ wtr,
                                                     float* __restrict__ wsq) {
    int idx = blockIdx.x * 256 + threadIdx.x;    // o*IC + i (exact grid)
    int o = idx >> 9, i = idx & (IC - 1);
    const float* p = w + (size_t)idx * KTAPS;
    float s = 0.f;
#pragma unroll
    for (int t = 0; t < KTAPS; ++t) {
        float v = p[t] * GAIN;
        wtr[(size_t)o * KTOT + t * IC + i] = v;
        s = fmaf(v, v, s);
    }
    wsq[idx] = s;
}

// ------------------------------------------------------------- prep: input
__global__ __launch_bounds__(256) void prep_x_kernel(const float* __restrict__ x,
                                                     const float* __restrict__ style,
                                                     float* __restrict__ xs) {
    int idx = blockIdx.x * 256 + threadIdx.x;    // exact grid
    int p  = idx % PHW;
    int bi = idx / PHW;
    int hh = p / PW, ww = p - hh * PW;
    float v = 0.f;
    if (hh >= 1 && hh <= HH && ww >= 1 && ww <= WW)
        v = x[(size_t)bi * HW + (hh - 1) * WW + (ww - 1)] * style[bi];
    xs[idx] = v;
}

// ------------------------------------------------------------- sigma
__global__ __launch_bounds__(256) void sigma_kernel(const float* __restrict__ wsq,
                                                    const float* __restrict__ style,
                                                    float* __restrict__ sigma) {
    int idx = blockIdx.x * 256 + threadIdx.x;    // b*OC + o
    int b = idx >> 9, o = idx & (OC - 1);
    const float* wr = wsq + (size_t)o * IC;
    const float* st = style + (size_t)b * IC;
    float acc = 0.f;
    for (int i = 0; i < IC; ++i) { float s = st[i]; acc = fmaf(wr[i], s * s, acc); }
    float xv = acc + EPS;
    float r = rsqrtf(xv);
    r = r * (1.5f - 0.5f * xv * r * r);
    sigma[idx] = r;
}

// ------------------------------------------------------------- main GEMM
static constexpr int BM = 64, BN = 256, BK = 16;
static constexpr int ASTR = 18;   // TDM pad: 16 dwords + 2  -> b64 frags conflict-free
static constexpr int BROW = 264;  // TDM pad: 256 dwords + 8 -> K-halves on disjoint banks

struct SM {
    float As[2][BM][ASTR];   //  9.2 KB  [m][k]
    float Bs[2][BK][BROW];   // 33.8 KB  [k][n] (n = 8 rows x 32 cols)
};

// Issue one TDM load: D# group0 built here, groups 1-3 passed in.
// clang-23 toolchain builtin arity: (u32x4, i32x8, i32x4, i32x4, i32x8, i32 cpol)
__device__ __forceinline__ void tdm_load(unsigned lds_addr, const void* gptr,
                                         i32x8 g1, i32x4 g2, i32x4 g3) {
    unsigned long long ga = (unsigned long long)(uintptr_t)gptr & ((1ull << 57) - 1);
    u32x4 g0;
    g0[0] = 1u;                                   // count=1, user descriptor
    g0[1] = lds_addr;                             // LDS byte address
    g0[2] = (unsigned)ga;                         // global addr [31:0]
    g0[3] = (unsigned)(ga >> 32) | (2u << 30);    // global addr [56:32] | type=2
    i32x8 gx;
    gx[0] = 0; gx[1] = 0; gx[2] = 0; gx[3] = 0;
    gx[4] = 0; gx[5] = 0; gx[6] = 0; gx[7] = 0;
    __builtin_amdgcn_tensor_load_to_lds(g0, g1, g2, g3, gx, 0);
}

__global__ __launch_bounds__(256) void conv_wmma_kernel(const float* __restrict__ xs,
                                                        const float* __restrict__ wtr,
                                                        const float* __restrict__ sigma,
                                                        float* __restrict__ out) {
    __shared__ SM sm;

    const int b   = blockIdx.z;
    const int ob  = blockIdx.y * BM;
    const int nb  = blockIdx.x * BN;              // 8 rows of 32
    const int h0  = nb >> 5;
    const int tid = threadIdx.x;

    const float* xsb = xs + (size_t)b * IC * PHW;
    const float* wob = wtr + (size_t)ob * KTOT;

    // wave / fragment maps (2M x 4N wave grid, wave tile 32M x 64N)
    const int lane = tid & 31;
    const int wid  = tid >> 5;
    const int m0   = (wid & 1) * 32;
    const int n0   = (wid >> 1) * 64;
    const int half = lane >> 4;                   // K-half (f32 16x16x4 layout)
    const int ml   = lane & 15;

    // ---- TDM descriptor groups 1-3 (loop-invariant) ----
    // A: 2D tile 16(x, contig) x 64(y, stride KTOT); pad 2 dw every 16 dw -> [64][18]
    i32x8 g1a;
    g1a[0] = (int)((2u << 16) | (1u << 20) | (3u << 22) | (1u << 25)); // 4B, pad on, 16dw->+2dw
    g1a[1] = (int)(4096u << 16);      // tensor_dim0 (generous; tile stays in-bounds)
    g1a[2] = (int)(4096u << 16);      // tensor_dim1
    g1a[3] = (int)(16u << 16);        // tile_dim0 = 16
    g1a[4] = 64;                      // tile_dim1 = 64, tile_dim2 = 0 (2D)
    g1a[5] = KTOT;                    // tensor_dim0_stride = 4608
    g1a[6] = 0;                       // strides hi
    g1a[7] = 0;
    // B: 3D tile 32(x) x 8(y, stride PW) x 16(z, stride PHW); pad 8 dw every 256 dw -> [16][264]
    i32x8 g1b;
    g1b[0] = (int)((2u << 16) | (1u << 20) | (7u << 22) | (7u << 25)); // 4B, pad on, 256dw->+8dw
    g1b[1] = (int)(4096u << 16);      // tensor_dim0
    g1b[2] = (int)(4096u << 16);      // tensor_dim1
    g1b[3] = (int)(32u << 16);        // tile_dim0 = 32
    g1b[4] = (int)(8u | (16u << 16)); // tile_dim1 = 8, tile_dim2 = 16
    g1b[5] = PW;                      // tensor_dim0_stride = 34
    g1b[6] = (int)((unsigned)PHW << 16); // tensor_dim1_stride lo16 = 1156
    g1b[7] = 0;
    i32x4 g2; g2[0] = 4096; g2[1] = 0; g2[2] = 0; g2[3] = 0;  // tensor_dim2 (generous)
    i32x4 g3; g3[0] = 0; g3[1] = 0; g3[2] = 0; g3[3] = 0;

    v8f acc[2][4] = {};

    // ---- prologue: DMA tile 0 into buffer 0 ----
    if (wid == 0) {
        tdm_load((unsigned)(uintptr_t)&sm.As[0][0][0], wob, g1a, g2, g3);
        tdm_load((unsigned)(uintptr_t)&sm.Bs[0][0][0], xsb, g1b, g2, g3); // k0=0: i0=0,kh=0,kw=0
        __builtin_amdgcn_s_wait_tensorcnt(0);
    }
    __syncthreads();

    int buf = 0;
    for (int k0 = 0; k0 < KTOT; k0 += BK) {
        // ---- DMA next tile into buf^1 (overlaps with compute below) ----
        if (wid == 0 && k0 + BK < KTOT) {
            const int k1  = k0 + BK;
            const int i0  = k1 & (IC - 1);
            const int tap = k1 >> 9;
            const int kh  = tap / 3;
            const int kw  = tap - kh * 3;
            tdm_load((unsigned)(uintptr_t)&sm.As[buf ^ 1][0][0], wob + k1, g1a, g2, g3);
            tdm_load((unsigned)(uintptr_t)&sm.Bs[buf ^ 1][0][0],
                     xsb + (size_t)i0 * PHW + (h0 + kh) * PW + kw, g1b, g2, g3);
        }

        // ---- 32 x V_WMMA_F32_16X16X4_F32 per wave per iteration ----
#pragma unroll
        for (int kk = 0; kk < BK; kk += 4) {
            const int ka = kk + half * 2;         // lanes 0-15: K,K+1 ; 16-31: K+2,K+3
            v2f a0 = *reinterpret_cast<const v2f*>(&sm.As[buf][m0 + ml][ka]);
            v2f a1 = *reinterpret_cast<const v2f*>(&sm.As[buf][m0 + 16 + ml][ka]);
#pragma unroll
            for (int s = 0; s < 4; ++s) {
                const int nn = n0 + s * 16 + ml;
                v2f bb;
                bb.x = sm.Bs[buf][ka][nn];
                bb.y = sm.Bs[buf][ka + 1][nn];
                acc[0][s] = __builtin_amdgcn_wmma_f32_16x16x4_f32(
                    false, a0, false, bb, (short)0, acc[0][s], false, false);
                acc[1][s] = __builtin_amdgcn_wmma_f32_16x16x4_f32(
                    false, a1, false, bb, (short)0, acc[1][s], false, false);
            }
        }

        if (wid == 0) __builtin_amdgcn_s_wait_tensorcnt(0);  // next tile landed
        __syncthreads();
        buf ^= 1;
    }

    // ---- epilogue: scale by sigma[b,o], store ----
#pragma unroll
    for (int mi = 0; mi < 2; ++mi) {
#pragma unroll
        for (int r = 0; r < 8; ++r) {
            const int o = ob + m0 + mi * 16 + half * 8 + r;
            const float sg = sigma[b * OC + o];
#pragma unroll
            for (int s = 0; s < 4; ++s) {
                out[((size_t)b * OC + o) * HW + nb + n0 + s * 16 + ml] = acc[mi][s][r] * sg;
            }
        }
    }
}

// ------------------------------------------------------------- launcher
extern "C" void kernel_launch(void* const* d_in, const int* in_sizes, int n_in,
                              void* d_out, int out_size, void* d_ws, size_t ws_size,
                              hipStream_t stream) {
    const float* x      = (const float*)d_in[0];
    const float* style  = (const float*)d_in[1];
    const float* weight = (const float*)d_in[2];
    float* out = (float*)d_out;

    float* xsb = (float*)d_ws;
    float* wtr = xsb + XS_N;
    float* wsq = wtr + WT_N;
    float* sig = wsq + WSQ_N;

    prep_w_kernel<<<(OC * IC) / 256, 256, 0, stream>>>(weight, wtr, wsq);
    prep_x_kernel<<<(int)(XS_N / 256), 256, 0, stream>>>(x, style, xsb);
    sigma_kernel<<<(NB * OC) / 256, 256, 0, stream>>>(wsq, style, sig);

    dim3 grid(HW / BN, OC / BM, NB);   // 4 x 8 x 16 = 512 blocks
    conv_wmma_kernel<<<grid, 256, 0, stream>>>(xsb, wtr, sig, out);
}